// GCN_46239617908904
// MI455X (gfx1250) — compile-verified
//
#include <hip/hip_runtime.h>
#include <math.h>

// GCN on MI455X (gfx1250, wave32).
// Phase 1: degree + rsqrt norms (native global_atomic_add_f32 via inline asm).
// Phase 2 (x3 layers): scatter-add of src-normalized features (L2-resident
//   accumulator, 12.8MB << 192MB L2; atomics resolved at L2 atomic units),
//   then fp32 WMMA GEMM (v_wmma_f32_16x16x4_f32) with fused dst-norm, bias,
//   ELU, residual.

#define D_FEAT 64
#define LDS_STRIDE 68   // 64 + 4 pad: avoids 64-bank stride conflicts, keeps float4 align

typedef float v2f __attribute__((ext_vector_type(2)));
typedef float v8f __attribute__((ext_vector_type(8)));

// Guaranteed-native FP32 atomic add (no CAS loop): CDNA5 GLOBAL_ATOMIC_ADD_F32,
// no-return form (STOREcnt), device scope so the RMW executes at the L2 atomic
// units and is coherent across all WGPs. s_endpgm's implicit wait-idle covers
// outstanding STOREcnt at kernel end.
__device__ __forceinline__ void atomic_add_f32(float* p, float v) {
    asm volatile("global_atomic_add_f32 %0, %1, off scope:SCOPE_DEV"
                 :
                 : "v"(p), "v"(v)
                 : "memory");
}

__global__ __launch_bounds__(256) void degree_kernel(const int* __restrict__ src,
                                                     const int* __restrict__ dst,
                                                     float* __restrict__ deg_out,
                                                     float* __restrict__ deg_in,
                                                     int E) {
    int e = blockIdx.x * 256 + threadIdx.x;
    if (e < E) {
        atomic_add_f32(&deg_out[src[e]], 1.0f);
        atomic_add_f32(&deg_in[dst[e]], 1.0f);
    }
}

__global__ __launch_bounds__(256) void norm_kernel(float* __restrict__ deg, int count) {
    int i = blockIdx.x * 256 + threadIdx.x;
    if (i < count) deg[i] = rsqrtf(fmaxf(deg[i], 1.0f));
}

// One thread per (edge, feature). Lanes cover consecutive features of one edge
// (coalesced gather + coalesced atomic scatter). src-side norm fused here.
__global__ __launch_bounds__(256) void scatter_kernel(const float* __restrict__ h,
                                                      const int* __restrict__ src,
                                                      const int* __restrict__ dst,
                                                      const float* __restrict__ norm_src,
                                                      float* __restrict__ agg,
                                                      int E) {
    unsigned gid = blockIdx.x * 256u + threadIdx.x;
    unsigned e = gid >> 6;
    int f = (int)(gid & 63u);
    if (e < (unsigned)E) {
        int s = src[e];
        int d = dst[e];
        float v = h[(size_t)s * D_FEAT + f] * norm_src[s];
        atomic_add_f32(&agg[(size_t)d * D_FEAT + f], v);
    }
}

// Block = 256 threads = 8 waves; block covers 32 nodes.
// Wave w: rows (w>>2)*16 .. +15 (local), cols (w&3)*16 .. +15.
// A = (agg * norm_dst) tile staged in LDS; B = W slice streamed from global.
// V_WMMA_F32_16X16X4_F32 layout:
//   A 16x4: lanes 0-15 -> M=lane, K={k0,k0+1}; lanes 16-31 -> M=lane-16, K={k0+2,k0+3}
//   B 4x16: lanes 0-15 -> N=lane, K={k0,k0+1}; lanes 16-31 -> N=lane-16, K={k0+2,k0+3}
//   C/D 16x16: VGPR r, lane l -> M = r + (l<16?0:8), N = l%16
__global__ __launch_bounds__(256) void gemm_wmma_kernel(const float* __restrict__ h_in,
                                                        const float* __restrict__ agg,
                                                        const float* __restrict__ norm_dst,
                                                        const float* __restrict__ W,
                                                        const float* __restrict__ bias,
                                                        float* __restrict__ out,
                                                        int n_nodes, int apply_elu) {
    __shared__ float a_lds[32 * LDS_STRIDE];

    const int node_base = blockIdx.x * 32;
    const int tid = threadIdx.x;

    // Cooperative load: 32 rows x 16 float4 = 512 chunks, 2 per thread.
    #pragma unroll
    for (int i = 0; i < 2; ++i) {
        int chunk = tid + i * 256;          // 0..511
        int row   = chunk >> 4;             // 0..31
        int f4    = (chunk & 15) * 4;       // 0,4,...,60
        int node  = node_base + row;
        float4 v = make_float4(0.f, 0.f, 0.f, 0.f);
        if (node < n_nodes) {
            v = *reinterpret_cast<const float4*>(agg + (size_t)node * D_FEAT + f4);
            float nd = norm_dst[node];
            v.x *= nd; v.y *= nd; v.z *= nd; v.w *= nd;
        }
        *reinterpret_cast<float4*>(&a_lds[row * LDS_STRIDE + f4]) = v;
    }
    __syncthreads();

    const int wave = tid >> 5;              // 0..7
    const int lane = tid & 31;
    const int row_tile = (wave >> 2) * 16;  // local row base: 0 or 16
    const int col_base = (wave & 3) * 16;   // 0,16,32,48
    const int half = lane >> 4;             // 0 or 1
    const int l15  = lane & 15;
    const int koff = half * 2;              // K sub-offset for this lane half
    const int ncol = col_base + l15;

    v8f c = {0.f, 0.f, 0.f, 0.f, 0.f, 0.f, 0.f, 0.f};
    const float* a_row = &a_lds[(row_tile + l15) * LDS_STRIDE];

    #pragma unroll
    for (int k0 = 0; k0 < D_FEAT; k0 += 4) {
        v2f a, b;
        a.x = a_row[k0 + koff];
        a.y = a_row[k0 + koff + 1];
        b.x = W[(size_t)(k0 + koff) * D_FEAT + ncol];
        b.y = W[(size_t)(k0 + koff + 1) * D_FEAT + ncol];
        // 8 args: (neg_a, A, neg_b, B, c_mod, C, reuse_a, reuse_b)
        c = __builtin_amdgcn_wmma_f32_16x16x4_f32(false, a, false, b,
                                                  (short)0, c, false, false);
    }

    const float bn = bias[ncol];
    #pragma unroll
    for (int r = 0; r < 8; ++r) {
        int m = node_base + row_tile + r + half * 8;
        if (m < n_nodes) {
            float t = c[r] + bn;
            if (apply_elu) t = (t > 0.f) ? t : expm1f(t);
            out[(size_t)m * D_FEAT + ncol] = h_in[(size_t)m * D_FEAT + ncol] + t;
        }
    }
}

extern "C" void kernel_launch(void* const* d_in, const int* in_sizes, int n_in,
                              void* d_out, int out_size, void* d_ws, size_t ws_size,
                              hipStream_t stream) {
    const float* x  = (const float*)d_in[0];
    const int* src  = (const int*)d_in[1];
    const int* dst  = (const int*)d_in[2];
    const float* W  = (const float*)d_in[3];   // [3][64][64]
    const float* b  = (const float*)d_in[4];   // [3][64]
    const int N = in_sizes[0] / D_FEAT;        // 50000
    const int E = in_sizes[1];                 // 800000

    float* ws       = (float*)d_ws;
    float* norm_src = ws;                      // N floats (deg_out -> norm in place)
    float* norm_dst = ws + N;                  // N floats (deg_in  -> norm in place)
    float* agg      = ws + 2 * (size_t)N;      // N*64 floats
    float* hbuf     = agg + (size_t)N * D_FEAT;// N*64 floats
    float* out      = (float*)d_out;

    // Degrees + norms.
    hipMemsetAsync(norm_src, 0, (size_t)2 * N * sizeof(float), stream);
    degree_kernel<<<(E + 255) / 256, 256, 0, stream>>>(src, dst, norm_src, norm_dst, E);
    norm_kernel<<<(2 * N + 255) / 256, 256, 0, stream>>>(norm_src, 2 * N);

    // 3 layers: x -> d_out -> hbuf -> d_out
    const float* h = x;
    float* layer_out[3] = { out, hbuf, out };
    for (int layer = 0; layer < 3; ++layer) {
        hipMemsetAsync(agg, 0, (size_t)N * D_FEAT * sizeof(float), stream);
        unsigned sthreads = (unsigned)E * (unsigned)D_FEAT;
        scatter_kernel<<<(sthreads + 255) / 256, 256, 0, stream>>>(h, src, dst,
                                                                   norm_src, agg, E);
        gemm_wmma_kernel<<<(N + 31) / 32, 256, 0, stream>>>(
            h, agg, norm_dst,
            W + (size_t)layer * D_FEAT * D_FEAT,
            b + (size_t)layer * D_FEAT,
            layer_out[layer], N, (layer < 2) ? 1 : 0);
        h = layer_out[layer];
    }
}